// NTXentLossChildrenFmapsPartwise_85624468013611
// MI455X (gfx1250) — compile-verified
//
#include <hip/hip_runtime.h>
#include <math.h>

typedef __attribute__((ext_vector_type(2))) float v2f;
typedef __attribute__((ext_vector_type(4))) float v4f;
typedef __attribute__((ext_vector_type(8))) float v8f;

#define WAVES 8
#define BLOCK_THREADS (WAVES * 32)
#define TEMP_INV 2.0f        // 1 / temperature, temperature = 0.5
#define NORM_EPS 1e-8f

// One block per (level, batch). Streams K, accumulates unnormalized Grams
// G_tr = t * r^T and G_tt = t * t^T with V_WMMA_F32_16X16X4_F32, plus per-row
// sum-of-squares in packed VALU; normalizes + masked logsumexp in the epilogue.
__global__ __launch_bounds__(BLOCK_THREADS)
void ntxent_level_kernel(const float* __restrict__ ts,
                         const float* __restrict__ rs,
                         int D,
                         float* __restrict__ partial)
{
    const int b    = blockIdx.x;
    const int tid  = threadIdx.x;
    // Wave id is wave-uniform: pin it to an SGPR so the compiler emits a
    // scalar loop (no EXEC-masked vector loop around the WMMAs).
    const int w    = __builtin_amdgcn_readfirstlane(tid >> 5);
    const int lane = tid & 31;
    const int m    = lane & 15;    // matrix row handled by this lane
    const int kh   = lane >> 4;    // K-half (lanes 16..31 hold upper K slots)

    const float* __restrict__ tp =
        ts + (size_t)b * 16 * D + (size_t)m * D + (w * 8 + 4 * kh);
    const float* __restrict__ rp =
        rs + (size_t)b * 16 * D + (size_t)m * D + (w * 8 + 4 * kh);

    // Two accumulators per Gram -> four independent WMMA chains.
    v8f gtr0 = {0.f, 0.f, 0.f, 0.f, 0.f, 0.f, 0.f, 0.f};
    v8f gtr1 = {0.f, 0.f, 0.f, 0.f, 0.f, 0.f, 0.f, 0.f};
    v8f gtt0 = {0.f, 0.f, 0.f, 0.f, 0.f, 0.f, 0.f, 0.f};
    v8f gtt1 = {0.f, 0.f, 0.f, 0.f, 0.f, 0.f, 0.f, 0.f};
    // Vector sum-of-squares accumulators: pk-FMA directly on the load
    // register pairs (no cross-operand register shuffles).
    v4f sst4 = {0.f, 0.f, 0.f, 0.f};
    v4f ssr4 = {0.f, 0.f, 0.f, 0.f};

    // Each wave owns K-slices of 8 columns, strided by WAVES*8 = 64.
    // Lane (m, kh) loads a contiguous float4 at its slice + 4*kh; the Gram is
    // invariant under K permutation, so the two K=4 WMMA steps use a
    // consistent (permuted) K mapping for both A and B operands.
    const int nsteps = D >> 6;     // D / (WAVES*8); scalar (D is an SGPR arg)
#pragma unroll 4
    for (int it = 0; it < nsteps; ++it) {
        v4f t4 = __builtin_nontemporal_load((const v4f*)tp);
        v4f r4 = __builtin_nontemporal_load((const v4f*)rp);
        tp += WAVES * 8;
        rp += WAVES * 8;

        v2f a0 = {t4.x, t4.y};
        v2f a1 = {t4.z, t4.w};
        v2f b0 = {r4.x, r4.y};
        v2f b1 = {r4.z, r4.w};

        // f32 16x16x4: A-layout(X) == B-layout(X^T), so the same regs feed
        // both operands for t*t^T, and identically-loaded r feeds B for t*r^T.
        gtr0 = __builtin_amdgcn_wmma_f32_16x16x4_f32(false, a0, false, b0,
                                                     (short)0, gtr0, false, false);
        gtt0 = __builtin_amdgcn_wmma_f32_16x16x4_f32(false, a0, false, a0,
                                                     (short)0, gtt0, false, false);
        gtr1 = __builtin_amdgcn_wmma_f32_16x16x4_f32(false, a1, false, b1,
                                                     (short)0, gtr1, false, false);
        gtt1 = __builtin_amdgcn_wmma_f32_16x16x4_f32(false, a1, false, a1,
                                                     (short)0, gtt1, false, false);

        // Co-executes with the XDL WMMAs on the VALU pipe (v_pk_fma_f32).
        sst4 += t4 * t4;
        ssr4 += r4 * r4;
    }

    const v8f gtr = gtr0 + gtr1;
    const v8f gtt = gtt0 + gtt1;
    const float sst = (sst4.x + sst4.y) + (sst4.z + sst4.w);
    const float ssr = (ssr4.x + ssr4.y) + (ssr4.z + ssr4.w);

    // ---- cross-wave reduction in LDS ----
    __shared__ float lgtr[WAVES][256];
    __shared__ float lgtt[WAVES][256];
    __shared__ float lsst[WAVES][32];
    __shared__ float lssr[WAVES][32];
    __shared__ float Gtr[256];
    __shared__ float Gtt[256];
    __shared__ float NT[16];
    __shared__ float NR[16];
    __shared__ float red[16];

    // C/D layout: VGPR g, lanes 0-15 -> (M=g, N=lane); lanes 16-31 -> (M=8+g, N=lane-16)
#pragma unroll
    for (int g = 0; g < 8; ++g) {
        const int M = g + 8 * kh;
        lgtr[w][M * 16 + m] = gtr[g];
        lgtt[w][M * 16 + m] = gtt[g];
    }
    lsst[w][lane] = sst;
    lssr[w][lane] = ssr;
    __syncthreads();

    {
        float str = 0.f, stt = 0.f;
#pragma unroll
        for (int w2 = 0; w2 < WAVES; ++w2) {
            str += lgtr[w2][tid];
            stt += lgtt[w2][tid];
        }
        Gtr[tid] = str;
        Gtt[tid] = stt;
    }
    if (tid < 16) {
        float at = 0.f, ar = 0.f;
#pragma unroll
        for (int w2 = 0; w2 < WAVES; ++w2) {
            at += lsst[w2][tid] + lsst[w2][tid + 16];
            ar += lssr[w2][tid] + lssr[w2][tid + 16];
        }
        NT[tid] = fmaxf(sqrtf(at), NORM_EPS);
        NR[tid] = fmaxf(sqrtf(ar), NORM_EPS);
    }
    __syncthreads();

    // ---- epilogue: row-wise logsumexp over 32 logits, minus positive ----
    if (tid < 16) {
        const int i = tid;
        const float nti = NT[i];
        float ltr[16], ltt[16];
        float mx = -INFINITY;
#pragma unroll
        for (int j = 0; j < 16; ++j) {
            const float s = Gtr[i * 16 + j] / (nti * NR[j]) * TEMP_INV;
            ltr[j] = s;
            mx = fmaxf(mx, s);
        }
#pragma unroll
        for (int j = 0; j < 16; ++j) {
            const float s = (j == i) ? -INFINITY
                                     : Gtt[i * 16 + j] / (nti * NT[j]) * TEMP_INV;
            ltt[j] = s;
            mx = fmaxf(mx, s);
        }
        float sum = 0.f;
#pragma unroll
        for (int j = 0; j < 16; ++j) {
            sum += __expf(ltr[j] - mx);
            sum += (j == i) ? 0.f : __expf(ltt[j] - mx);
        }
        const float lse = mx + __logf(sum);
        red[i] = lse - ltr[i];   // ltr[i] is the positive logit
    }
    __syncthreads();

    if (tid == 0) {
        float s = 0.f;
#pragma unroll
        for (int j = 0; j < 16; ++j) s += red[j];
        partial[b] = s;
    }
}

// Deterministic final reduction (fixed serial order; trivial cost: 768 adds).
__global__ void ntxent_reduce_kernel(const float* __restrict__ partial, int n,
                                     float* __restrict__ out)
{
    if (blockIdx.x == 0 && threadIdx.x == 0) {
        float s = 0.f;
        for (int i = 0; i < n; ++i) s += partial[i];
        out[0] = s * (1.0f / 512.0f);   // / (2 * batch_size)
    }
}

extern "C" void kernel_launch(void* const* d_in, const int* in_sizes, int n_in,
                              void* d_out, int out_size, void* d_ws, size_t ws_size,
                              hipStream_t stream)
{
    const float* ts0 = (const float*)d_in[0];
    const float* rs0 = (const float*)d_in[1];
    const float* ts1 = (const float*)d_in[2];
    const float* rs1 = (const float*)d_in[3];
    const float* ts2 = (const float*)d_in[4];
    const float* rs2 = (const float*)d_in[5];
    // d_in[6] (partnet_ids) is unused by the reference loss.

    const int B = in_sizes[0] / (16 * 4096);   // 256 for the reference shapes
    float* partial = (float*)d_ws;             // 3*B floats of scratch

    ntxent_level_kernel<<<B, BLOCK_THREADS, 0, stream>>>(ts0, rs0, 4096, partial);
    ntxent_level_kernel<<<B, BLOCK_THREADS, 0, stream>>>(ts1, rs1, 2048, partial + B);
    ntxent_level_kernel<<<B, BLOCK_THREADS, 0, stream>>>(ts2, rs2, 1024, partial + 2 * B);
    ntxent_reduce_kernel<<<1, 32, 0, stream>>>(partial, 3 * B, (float*)d_out);
}